// MVAU_54606214201638
// MI455X (gfx1250) — compile-verified
//
#include <hip/hip_runtime.h>

#define BB 8
#define CC 96
#define LL 384
#define KK 3
#define EPSF 1e-6f

typedef float v2f __attribute__((ext_vector_type(2)));
typedef float v8f __attribute__((ext_vector_type(8)));

__device__ __forceinline__ v8f wmma_f32_k4(v2f a, v2f b, v8f c) {
  // V_WMMA_F32_16X16X4_F32: D = A(16x4) * B(4x16) + C(16x16), fp32
  return __builtin_amdgcn_wmma_f32_16x16x4_f32(false, a, false, b, (short)0, c, false, false);
}

// CDNA5 async global->LDS copy (ASYNCcnt-tracked). ldsOff = LDS byte address
// (low 32 bits of a generic pointer to __shared__), gaddr = global address.
__device__ __forceinline__ void async_g2l_b32(unsigned int ldsOff, const float* gaddr) {
  asm volatile("global_load_async_to_lds_b32 %0, %1, off"
               :: "v"(ldsOff), "v"(gaddr) : "memory");
}
__device__ __forceinline__ void wait_async0() {
  asm volatile("s_wait_asynccnt 0" ::: "memory");
}

// ---------------- LayerNorm over channel dim ----------------
__global__ void k_ln(const float* __restrict__ x, const float* __restrict__ lnw,
                     const float* __restrict__ lnb, float* __restrict__ xn) {
  int col = blockIdx.x * blockDim.x + threadIdx.x;
  if (col >= BB * LL) return;
  int b = col / LL, l = col % LL;
  const float* xb = x + (size_t)b * CC * LL + l;
  float s = 0.f, s2 = 0.f;
  for (int c = 0; c < CC; ++c) { float v = xb[c * LL]; s += v; s2 += v * v; }
  float u = s / CC;
  float var = s2 / CC - u * u;
  float inv = rsqrtf(var + EPSF);
  float* xo = xn + (size_t)b * CC * LL + l;
  for (int c = 0; c < CC; ++c) {
    float v = xb[c * LL];
    xo[c * LL] = lnw[c] * ((v - u) * inv) + lnb[c];
  }
}

// ---------------- y = wp1 @ xn + bp1  ->  a, z  (WMMA fp32) ----------------
__global__ void k_gemm1(const float* __restrict__ xn, const float* __restrict__ wp1,
                        const float* __restrict__ bp1, float* __restrict__ aB,
                        float* __restrict__ zB) {
  const int DT = (2 * CC) / 16;  // 12
  const int LT = LL / 16;        // 24
  int wave = threadIdx.x >> 5;
  int lane = threadIdx.x & 31;
  int tile = blockIdx.x * 4 + wave;
  if (tile >= BB * DT * LT) return;
  int b  = tile / (DT * LT);
  int r  = tile % (DT * LT);
  int dt = r / LT, lt = r % LT;
  int d0 = dt * 16, l0 = lt * 16;
  int m = lane & 15;
  int half = lane >> 4;
  int kBase = half * 2;
  const float* xb = xn + (size_t)b * CC * LL;
  v8f acc = {};
  for (int c0 = 0; c0 < CC; c0 += 4) {
    v2f av, bv;
    av.x = wp1[(d0 + m) * CC + c0 + kBase];
    av.y = wp1[(d0 + m) * CC + c0 + kBase + 1];
    bv.x = xb[(c0 + kBase) * LL + l0 + m];
    bv.y = xb[(c0 + kBase + 1) * LL + l0 + m];
    acc = wmma_f32_k4(av, bv, acc);
  }
  int col = l0 + m;
#pragma unroll
  for (int rI = 0; rI < 8; ++rI) {
    int d = d0 + rI + half * 8;
    float v = acc[rI] + bp1[d];
    if (d < CC) aB[((size_t)b * CC + d) * LL + col] = v;
    else        zB[((size_t)b * CC + (d - CC)) * LL + col] = v;
  }
}

// ---------------- enc1 = mean_l(a) @ W1^T + b1, plus row max/min ----------------
__global__ void k_enc1(const float* __restrict__ a, const float* __restrict__ W1,
                       const float* __restrict__ b1, float* __restrict__ enc1,
                       float* __restrict__ enc1mm) {
  __shared__ float ml[CC];
  __shared__ float ev[CC];
  int b = blockIdx.x, t = threadIdx.x;  // 96 threads
  const float* row = a + ((size_t)b * CC + t) * LL;
  float s = 0.f;
  for (int l = 0; l < LL; ++l) s += row[l];
  ml[t] = s / LL;
  __syncthreads();
  float acc = b1[t];
  for (int c = 0; c < CC; ++c) acc += ml[c] * W1[t * CC + c];
  enc1[b * CC + t] = acc;
  ev[t] = acc;
  __syncthreads();
  if (t == 0) {
    float mx = ev[0], mn = ev[0];
    for (int i = 1; i < CC; ++i) { mx = fmaxf(mx, ev[i]); mn = fminf(mn, ev[i]); }
    enc1mm[b * 2] = mx; enc1mm[b * 2 + 1] = mn;
  }
}

// ---------------- enc2 = mean_c(a) @ W2^T + b2, plus row max/min ----------------
__global__ void k_enc2(const float* __restrict__ a, const float* __restrict__ W2,
                       const float* __restrict__ b2, float* __restrict__ enc2,
                       float* __restrict__ enc2mm) {
  __shared__ float mc[LL];
  __shared__ float ev[LL];
  int b = blockIdx.x, t = threadIdx.x;  // 384 threads
  const float* ab = a + (size_t)b * CC * LL;
  float s = 0.f;
  for (int c = 0; c < CC; ++c) s += ab[c * LL + t];
  mc[t] = s / CC;
  __syncthreads();
  float acc = b2[t];
  for (int l = 0; l < LL; ++l) acc += mc[l] * W2[t * LL + l];
  enc2[b * LL + t] = acc;
  ev[t] = acc;
  __syncthreads();
  if (t == 0) {
    float mx = ev[0], mn = ev[0];
    for (int i = 1; i < LL; ++i) { mx = fmaxf(mx, ev[i]); mn = fminf(mn, ev[i]); }
    enc2mm[b * 2] = mx; enc2mm[b * 2 + 1] = mn;
  }
}

// ---------------- enc3 = mean_l(a^T) @ W3^T + b3, plus row max/min ----------------
__global__ void k_enc3(const float* __restrict__ a, const float* __restrict__ W3,
                       const float* __restrict__ b3, float* __restrict__ enc3,
                       float* __restrict__ enc3mm) {
  __shared__ float ml[BB];
  __shared__ float ev[BB];
  int c = blockIdx.x;
  int t = threadIdx.x;          // 256 threads = 8 waves; wave w handles batch b=w
  int w = t >> 5, lane = t & 31;
  float s = 0.f;
  const float* row = a + ((size_t)w * CC + c) * LL;
  for (int l = lane; l < LL; l += 32) s += row[l];
  for (int off = 16; off > 0; off >>= 1) s += __shfl_xor(s, off, 32);
  if (lane == 0) ml[w] = s / LL;
  __syncthreads();
  if (t < BB) {
    float acc = b3[t];
    for (int b = 0; b < BB; ++b) acc += ml[b] * W3[t * BB + b];
    enc3[c * BB + t] = acc;
    ev[t] = acc;
  }
  __syncthreads();
  if (t == 0) {
    float mx = ev[0], mn = ev[0];
    for (int i = 1; i < BB; ++i) { mx = fmaxf(mx, ev[i]); mn = fminf(mn, ev[i]); }
    enc3mm[c * 2] = mx; enc3mm[c * 2 + 1] = mn;
  }
}

// ---------------- au1: att = s1 * aug1  (rank-1 softmax, lane-per-p) ----------------
__global__ void k_au1(const float* __restrict__ a, const float* __restrict__ enc1,
                      const float* __restrict__ enc1mm, const float* __restrict__ s1p,
                      float* __restrict__ att) {
  __shared__ float f[CC];
  __shared__ float e[CC];
  int n = blockIdx.x;               // n in [0, B*L)
  int b = n / LL, l = n % LL, eb = n % BB;
  int t = threadIdx.x;              // 96 threads: p = t
  // Stage f and e straight into LDS with CDNA5 async global->LDS copies.
  async_g2l_b32((unsigned int)(size_t)&f[t], a + ((size_t)b * CC + t) * LL + l);
  async_g2l_b32((unsigned int)(size_t)&e[t], enc1 + eb * CC + t);
  wait_async0();
  __syncthreads();
  float s = f[t];
  float m = (s >= 0.f) ? s * enc1mm[eb * 2] : s * enc1mm[eb * 2 + 1];
  float num = 0.f, den = 0.f;
  for (int q = 0; q < CC; ++q) {
    float wv = __expf(s * e[q] - m);
    den += wv; num += wv * f[q];
  }
  att[((size_t)b * CC + t) * LL + l] = s1p[0] * (num / den);
}

// ---------------- au2: att += s2 * aug2 ----------------
__global__ void k_au2(const float* __restrict__ a, const float* __restrict__ enc2,
                      const float* __restrict__ enc2mm, const float* __restrict__ s2p,
                      float* __restrict__ att) {
  __shared__ float f[LL];
  __shared__ float e[LL];
  int n = blockIdx.x;               // n in [0, B*C)
  int b = n / CC, c = n % CC, eb = n % BB;
  int t = threadIdx.x;              // 384 threads: p = t = l
  size_t base = ((size_t)b * CC + c) * LL;
  async_g2l_b32((unsigned int)(size_t)&f[t], a + base + t);
  async_g2l_b32((unsigned int)(size_t)&e[t], enc2 + eb * LL + t);
  wait_async0();
  __syncthreads();
  float s = f[t];
  float m = (s >= 0.f) ? s * enc2mm[eb * 2] : s * enc2mm[eb * 2 + 1];
  float num = 0.f, den = 0.f;
  for (int q = 0; q < LL; ++q) {
    float wv = __expf(s * e[q] - m);
    den += wv; num += wv * f[q];
  }
  att[base + t] += s2p[0] * (num / den);
}

// ---------------- au3: att += s3 * aug3, then fuse g = z * att in place ----------------
__global__ void k_au3_g(const float* __restrict__ a, const float* __restrict__ z,
                        const float* __restrict__ enc3, const float* __restrict__ enc3mm,
                        const float* __restrict__ s3p, float* __restrict__ att) {
  int n = blockIdx.x * blockDim.x + threadIdx.x;   // n = c*LL + l in [0, C*L)
  if (n >= CC * LL) return;
  int c = n / LL, l = n % LL;
  int er = n % CC;
  float e[BB], f[BB];
#pragma unroll
  for (int q = 0; q < BB; ++q) e[q] = enc3[er * BB + q];
#pragma unroll
  for (int p = 0; p < BB; ++p) f[p] = a[((size_t)p * CC + c) * LL + l];
  float emx = enc3mm[er * 2], emn = enc3mm[er * 2 + 1];
  float s3 = s3p[0];
#pragma unroll
  for (int p = 0; p < BB; ++p) {
    float s = f[p];
    float m = (s >= 0.f) ? s * emx : s * emn;
    float num = 0.f, den = 0.f;
#pragma unroll
    for (int q = 0; q < BB; ++q) {
      float wv = __expf(s * e[q] - m);
      den += wv; num += wv * f[q];
    }
    size_t idx = ((size_t)p * CC + c) * LL + l;
    float av = att[idx] + s3 * (num / den);
    att[idx] = z[idx] * av;   // g stored in place of att
  }
}

// ---------------- conv1d(g, wp2) + ReLU*scale + shortcut  (WMMA fp32) ----------------
__global__ void k_conv(const float* __restrict__ g, const float* __restrict__ wp2,
                       const float* __restrict__ scale, const float* __restrict__ x,
                       float* __restrict__ out) {
  const int DT = CC / 16;  // 6
  const int LT = LL / 16;  // 24
  int wave = threadIdx.x >> 5;
  int lane = threadIdx.x & 31;
  int tile = blockIdx.x * 4 + wave;
  if (tile >= BB * DT * LT) return;
  int b  = tile / (DT * LT);
  int r  = tile % (DT * LT);
  int dt = r / LT, lt = r % LT;
  int d0 = dt * 16, l0 = lt * 16;
  int m = lane & 15;
  int half = lane >> 4;
  int kBase = half * 2;
  const float* gb = g + (size_t)b * CC * LL;
  v8f acc = {};
  for (int k = 0; k < KK; ++k) {
    int col = l0 + m + k - 1;
    bool ok = (col >= 0) && (col < LL);
    for (int c0 = 0; c0 < CC; c0 += 4) {
      v2f av, bv;
      av.x = wp2[((d0 + m) * CC + c0 + kBase) * KK + k];
      av.y = wp2[((d0 + m) * CC + c0 + kBase + 1) * KK + k];
      bv.x = ok ? gb[(c0 + kBase) * LL + col] : 0.f;
      bv.y = ok ? gb[(c0 + kBase + 1) * LL + col] : 0.f;
      acc = wmma_f32_k4(av, bv, acc);
    }
  }
  int col = l0 + m;
#pragma unroll
  for (int rI = 0; rI < 8; ++rI) {
    int d = d0 + rI + half * 8;
    float zc = acc[rI];
    out[((size_t)b * CC + d) * LL + col] =
        fmaxf(zc, 0.f) * scale[d] + x[((size_t)b * CC + d) * LL + col];
  }
}

extern "C" void kernel_launch(void* const* d_in, const int* in_sizes, int n_in,
                              void* d_out, int out_size, void* d_ws, size_t ws_size,
                              hipStream_t stream) {
  (void)in_sizes; (void)n_in; (void)out_size; (void)ws_size;
  const float* x    = (const float*)d_in[0];
  const float* lnw  = (const float*)d_in[1];
  const float* lnb  = (const float*)d_in[2];
  const float* scl  = (const float*)d_in[3];   // (1,C,1) -> 96 values
  const float* s1   = (const float*)d_in[4];
  const float* s2   = (const float*)d_in[5];
  const float* s3   = (const float*)d_in[6];
  const float* W1   = (const float*)d_in[7];
  const float* b1   = (const float*)d_in[8];
  const float* W2   = (const float*)d_in[9];
  const float* b2   = (const float*)d_in[10];
  const float* W3   = (const float*)d_in[11];
  const float* b3   = (const float*)d_in[12];
  const float* wp1  = (const float*)d_in[13];
  const float* bp1  = (const float*)d_in[14];
  const float* wp2  = (const float*)d_in[15];
  float* out = (float*)d_out;

  const size_t NTOT = (size_t)BB * CC * LL;    // 294912 floats
  float* ws     = (float*)d_ws;
  float* xn     = ws;
  float* aB     = xn + NTOT;
  float* zB     = aB + NTOT;
  float* att    = zB + NTOT;
  float* enc1   = att + NTOT;
  float* enc1mm = enc1 + BB * CC;
  float* enc2   = enc1mm + 2 * BB;
  float* enc2mm = enc2 + BB * LL;
  float* enc3   = enc2mm + 2 * BB;
  float* enc3mm = enc3 + CC * BB;

  k_ln   <<<(BB * LL + 255) / 256, 256, 0, stream>>>(x, lnw, lnb, xn);
  k_gemm1<<<(BB * 12 * 24 + 3) / 4, 128, 0, stream>>>(xn, wp1, bp1, aB, zB);
  k_enc1 <<<BB, CC, 0, stream>>>(aB, W1, b1, enc1, enc1mm);
  k_enc2 <<<BB, LL, 0, stream>>>(aB, W2, b2, enc2, enc2mm);
  k_enc3 <<<CC, 256, 0, stream>>>(aB, W3, b3, enc3, enc3mm);
  k_au1  <<<BB * LL, CC, 0, stream>>>(aB, enc1, enc1mm, s1, att);
  k_au2  <<<BB * CC, LL, 0, stream>>>(aB, enc2, enc2mm, s2, att);
  k_au3_g<<<(CC * LL + 255) / 256, 256, 0, stream>>>(aB, zB, enc3, enc3mm, s3, att);
  k_conv <<<(BB * 6 * 24 + 3) / 4, 128, 0, stream>>>(att, wp2, scl, x, out);
}